// InputEmbeddings_12970801234299
// MI455X (gfx1250) — compile-verified
//
#include <hip/hip_runtime.h>
#include <hip/hip_bf16.h>

// ---------------------------------------------------------------------------
// InputEmbeddings for MI455X (gfx1250)
//   B=256, S=128, d=1024, dq=256
//   out = type_emb[t] + pos_emb[idx] + final_pos_emb[s]
//         + (t==1) * (relu(sv@W1+b1)@W2 + b2 + pc@Tw + tb)
//   mask = (t == 0)
// Memory-bound on the 134MB output store; GEMMs done exactly in f32 via
// V_WMMA_F32_16X16X4_F32.
// ---------------------------------------------------------------------------

typedef __attribute__((ext_vector_type(2))) float v2f;
typedef __attribute__((ext_vector_type(8))) float v8f;

#define D_MODEL 1024
#define D_Q     256
#define BATCH   256
#define SEQ     128
#define K_TEXT  768

// ---------------------------------------------------------------------------
// Kernel A: h[m][q] = relu( sum_{k<5} sv[m][k] * w1[k][q] + b1[q] )
//   grid: (BATCH) blocks x (D_Q) threads
// ---------------------------------------------------------------------------
__global__ __launch_bounds__(D_Q) void style_hidden_kernel(
    const float* __restrict__ sv,   // [256,5]
    const float* __restrict__ w1,   // [5,256]
    const float* __restrict__ b1,   // [256]
    float* __restrict__ h)          // [256,256]
{
    const int m = blockIdx.x;
    const int q = threadIdx.x;
    const float* svr = sv + m * 5;
    float acc = b1[q];
    acc += svr[0] * w1[0 * D_Q + q];
    acc += svr[1] * w1[1 * D_Q + q];
    acc += svr[2] * w1[2 * D_Q + q];
    acc += svr[3] * w1[3 * D_Q + q];
    acc += svr[4] * w1[4 * D_Q + q];
    h[m * D_Q + q] = fmaxf(acc, 0.0f);
}

// ---------------------------------------------------------------------------
// Kernel B: combined[256,1024] = h @ W2 + P @ Tw + b2 + tb
//   WMMA f32 16x16x4. One wave -> one 16x16 C tile. 8 waves/block cover a
//   16(M) x 128(N) strip. grid = (1024/128, 256/16) = (8,16)
//
//   A fragment (16x4 f32, 2 VGPRs/lane):
//     lanes 0-15:  M=lane,   K = k, k+1
//     lanes 16-31: M=lane-16,K = k+2, k+3
//   B fragment (4x16 f32, 2 VGPRs/lane): mirrored (lane = N, halves = K pairs)
//   C/D (16x16 f32, 8 VGPRs): VGPR r -> M=r (lanes 0-15), M=r+8 (lanes 16-31)
// ---------------------------------------------------------------------------
__global__ __launch_bounds__(256) void combined_gemm_kernel(
    const float* __restrict__ h,    // [256,256]   (M x K1)
    const float* __restrict__ w2,   // [256,1024]  (K1 x N)
    const float* __restrict__ b2,   // [1024]
    const float* __restrict__ pc,   // [256,768]   (M x K2)
    const float* __restrict__ tw,   // [768,1024]  (K2 x N)
    const float* __restrict__ tb,   // [1024]
    float* __restrict__ out)        // [256,1024]
{
    const int wave = threadIdx.x >> 5;        // 0..7
    const int lane = threadIdx.x & 31;
    const int half = lane >> 4;               // 0 or 1
    const int lid  = lane & 15;
    const int m0 = blockIdx.y * 16;
    const int n0 = blockIdx.x * 128 + wave * 16;

    v8f c = {};

    // ---- style path: K = 256 ----
    {
        const float* arow = h + (m0 + lid) * D_Q;   // A row for this lane
        const float* bcol = w2 + n0 + lid;          // B column for this lane
        for (int k = 0; k < D_Q; k += 4) {
            const int ka = k + 2 * half;
            v2f a, b;
            // contiguous pair -> 8B load
            const v2f av = *(const v2f*)(arow + ka);
            a = av;
            b.x = bcol[(ka + 0) * D_MODEL];
            b.y = bcol[(ka + 1) * D_MODEL];
            c = __builtin_amdgcn_wmma_f32_16x16x4_f32(
                    false, a, false, b, (short)0, c, false, false);
        }
    }

    // ---- text path: K = 768 ----
    {
        const float* arow = pc + (m0 + lid) * K_TEXT;
        const float* bcol = tw + n0 + lid;
        for (int k = 0; k < K_TEXT; k += 4) {
            const int ka = k + 2 * half;
            v2f a, b;
            const v2f av = *(const v2f*)(arow + ka);
            a = av;
            b.x = bcol[(ka + 0) * D_MODEL];
            b.y = bcol[(ka + 1) * D_MODEL];
            c = __builtin_amdgcn_wmma_f32_16x16x4_f32(
                    false, a, false, b, (short)0, c, false, false);
        }
    }

    const float bias = b2[n0 + lid] + tb[n0 + lid];
    #pragma unroll
    for (int r = 0; r < 8; ++r) {
        out[(m0 + r + 8 * half) * D_MODEL + (n0 + lid)] = c[r] + bias;
    }
}

// ---------------------------------------------------------------------------
// Kernel C: final assembly (bandwidth-bound; float4 vectorized)
//   grid: B*S blocks x 256 threads; each block emits one 1024-float row
// ---------------------------------------------------------------------------
__global__ __launch_bounds__(256) void assemble_kernel(
    const int*   __restrict__ types,     // [256,128]
    const int*   __restrict__ idxs,      // [256,128]
    const float* __restrict__ type_emb,  // [5,1024]
    const float* __restrict__ pos_emb,   // [128,1024]
    const float* __restrict__ final_pe,  // [128,1024]
    const float* __restrict__ combined,  // [256,1024]
    float* __restrict__ out_emb,         // [256,128,1024]
    float* __restrict__ out_mask)        // [256,128]
{
    const int bs = blockIdx.x;           // 0 .. B*S-1
    const int b  = bs >> 7;
    const int s  = bs & (SEQ - 1);
    const int t  = types[bs];
    const int pi = idxs[bs];

    const float4* te = (const float4*)(type_emb + (size_t)t  * D_MODEL);
    const float4* pe = (const float4*)(pos_emb  + (size_t)pi * D_MODEL);
    const float4* fe = (const float4*)(final_pe + (size_t)s  * D_MODEL);
    const float4* cb = (const float4*)(combined + (size_t)b  * D_MODEL);
    float4*       o  = (float4*)(out_emb + (size_t)bs * D_MODEL);

    const int j = threadIdx.x;           // 0..255 -> 4 floats each
    float4 v = te[j];
    float4 p = pe[j];
    float4 f = fe[j];
    float4 r;
    r.x = v.x + p.x + f.x;
    r.y = v.y + p.y + f.y;
    r.z = v.z + p.z + f.z;
    r.w = v.w + p.w + f.w;
    if (t == 1) {
        float4 cv = cb[j];
        r.x += cv.x; r.y += cv.y; r.z += cv.z; r.w += cv.w;
    }
    o[j] = r;

    if (j == 0) {
        out_mask[bs] = (t == 0) ? 1.0f : 0.0f;
    }
}

// ---------------------------------------------------------------------------
// kernel_launch
//   d_in order (setup_inputs): 0 element_types, 1 element_indices,
//   2 style_vectors, 3 page_captions, 4 type_emb, 5 pos_emb, 6 style_w1,
//   7 style_b1, 8 style_w2, 9 style_b2, 10 text_w, 11 text_b, 12 final_pos_emb
//   d_out: [B*S*d] emb floats, then [B*S] mask values
// ---------------------------------------------------------------------------
extern "C" void kernel_launch(void* const* d_in, const int* in_sizes, int n_in,
                              void* d_out, int out_size, void* d_ws, size_t ws_size,
                              hipStream_t stream) {
    const int*   types    = (const int*)  d_in[0];
    const int*   idxs     = (const int*)  d_in[1];
    const float* sv       = (const float*)d_in[2];
    const float* pc       = (const float*)d_in[3];
    const float* type_emb = (const float*)d_in[4];
    const float* pos_emb  = (const float*)d_in[5];
    const float* w1       = (const float*)d_in[6];
    const float* b1       = (const float*)d_in[7];
    const float* w2       = (const float*)d_in[8];
    const float* b2       = (const float*)d_in[9];
    const float* tw       = (const float*)d_in[10];
    const float* tbv      = (const float*)d_in[11];
    const float* final_pe = (const float*)d_in[12];

    float* ws       = (float*)d_ws;
    float* h        = ws;                         // [256,256]   = 256 KB
    float* combined = ws + BATCH * D_Q;           // [256,1024]  = 1 MB

    float* out_emb  = (float*)d_out;                          // B*S*d floats
    float* out_mask = out_emb + (size_t)BATCH * SEQ * D_MODEL; // B*S values

    // A: hidden style layer (K=5, trivial)
    style_hidden_kernel<<<dim3(BATCH), dim3(D_Q), 0, stream>>>(sv, w1, b1, h);

    // B: combined = h@W2 + pc@Tw + b2 + tb  (WMMA f32 16x16x4)
    combined_gemm_kernel<<<dim3(D_MODEL / 128, BATCH / 16), dim3(256), 0, stream>>>(
        h, w2, b2, pc, tw, tbv, combined);

    // C: assemble output rows + padding mask
    assemble_kernel<<<dim3(BATCH * SEQ), dim3(256), 0, stream>>>(
        types, idxs, type_emb, pos_emb, final_pe, combined, out_emb, out_mask);
}